// RetinaNet_16140487098777
// MI455X (gfx1250) — compile-verified
//
#include <hip/hip_runtime.h>
#include <cstdint>

// ---- problem constants (match reference) ----
#define NCLS 81      // classes incl. background
#define NFG  80      // foreground classes
#define MAXO 100     // MAX_OUT per class
constexpr int   TILE = 128;    // anchors per block in decode kernel
constexpr int   CAP  = 8192;   // per-class candidate capacity in scratch
constexpr float IOU_THR = 0.45f;

// output layout (floats): class_id[80*100] | sel_s[80*100] | sel_b[80*100*4] | oks[80*100]
constexpr int OFF_CID = 0;
constexpr int OFF_S   = NFG * MAXO;          // 8000
constexpr int OFF_B   = 2 * NFG * MAXO;      // 16000
constexpr int OFF_OK  = OFF_B + 4 * NFG * MAXO; // 48000

// ------------------------------------------------------------------
// Kernel 0: zero the per-class candidate counters (must be done every
// call; harness does not re-initialize scratch between replays).
// ------------------------------------------------------------------
__global__ void rn_zero_counts(int* __restrict__ counts) {
  if (threadIdx.x < NFG) counts[threadIdx.x] = 0;
}

// ------------------------------------------------------------------
// Kernel 1: stream cla via async global->LDS, softmax-max/sum per
// anchor, decode box, append (at most one) candidate per anchor.
// Since softmax probs sum to 1, only the argmax class can exceed 0.5.
// ------------------------------------------------------------------
__global__ __launch_bounds__(TILE) void rn_decode_extract(
    const float* __restrict__ cla, const float* __restrict__ reg,
    const float* __restrict__ anc, float* __restrict__ sws,
    float4* __restrict__ bws, int* __restrict__ counts, int nAnch) {
  __shared__ __align__(16) float tile[TILE * NCLS];  // 41,472 B

  const int tid = threadIdx.x;
  const int a0  = blockIdx.x * TILE;

  // ---- CDNA5 async copy: cla[a0*81 .. (a0+TILE)*81) -> LDS ----
  // 16B chunks: TILE*81*4/16 = 2592, base is 16B aligned (a0 % 128 == 0).
  const unsigned lds0 = (unsigned)(uintptr_t)(void*)tile;  // low 32b = LDS byte addr
  const uint64_t g0   = (uint64_t)(uintptr_t)(cla + (size_t)a0 * NCLS);
  const int chunks = TILE * NCLS * 4 / 16;
  for (int i = tid; i < chunks; i += TILE) {
    unsigned la = lds0 + (unsigned)i * 16u;  // LDS dest byte address
    unsigned va = (unsigned)i * 16u;         // 32-bit byte offset vs SADDR base
    asm volatile("global_load_async_to_lds_b128 %0, %1, %2"
                 :: "v"(la), "v"(va), "s"(g0) : "memory");
  }
  asm volatile("s_wait_asynccnt 0" ::: "memory");
  __syncthreads();

  const int a = a0 + tid;
  if (a >= nAnch) return;

  // ---- softmax stats over this anchor's row (from LDS) ----
  const float* row = tile + tid * NCLS;
  float m = row[0];
  int   am = 0;
  for (int c = 1; c < NCLS; ++c) {
    float v = row[c];
    if (v > m) { m = v; am = c; }
  }
  float s = 0.f;
  for (int c = 0; c < NCLS; ++c) s += __expf(row[c] - m);

  // p_argmax = exp(0)/s = 1/s ; it is the only prob that can exceed 0.5
  if (am != 0 && s < 2.0f) {
    float score = 1.0f / s;
    float4 an = ((const float4*)anc)[a];  // y1 x1 y2 x2
    float4 rg = ((const float4*)reg)[a];
    float ah  = an.z - an.x;
    float aw  = an.w - an.y;
    float acy = (an.x + an.z) * 0.5f;
    float acx = (an.y + an.w) * 0.5f;
    float cy = rg.x * 0.1f * ah + acy;   // /10
    float cx = rg.y * 0.1f * aw + acx;   // /10
    float h  = __expf(rg.z * 0.2f) * ah; // /5
    float w  = __expf(rg.w * 0.2f) * aw; // /5
    int cls = am - 1;
    int pos = atomicAdd(&counts[cls], 1);
    if (pos < CAP) {
      sws[(size_t)cls * CAP + pos] = score;
      bws[(size_t)cls * CAP + pos] =
          make_float4(cy - 0.5f * h, cx - 0.5f * w, cy + 0.5f * h, cx + 0.5f * w);
    }
  }
}

// ------------------------------------------------------------------
// Kernel 2: per-class greedy NMS. One block per class. Scores in LDS
// (suppressed -> 0); boxes stay in L2-resident scratch.
// ------------------------------------------------------------------
__global__ __launch_bounds__(256) void rn_nms(
    const float* __restrict__ sws, const float4* __restrict__ bws,
    const int* __restrict__ counts, float* __restrict__ out) {
  __shared__ float  sc[CAP];    // 32 KB
  __shared__ float  rs[256];
  __shared__ int    ri[256];
  __shared__ float4 selb;
  __shared__ int    seli;

  const int c   = blockIdx.x;   // foreground class index 0..79 (class id c+1)
  const int tid = threadIdx.x;
  const int K   = min(counts[c], CAP);
  const float4* bx = bws + (size_t)c * CAP;

  for (int i = tid; i < K; i += 256) sc[i] = sws[(size_t)c * CAP + i];
  __syncthreads();

  for (int t = 0; t < MAXO; ++t) {
    // ---- block-wide argmax over surviving candidates ----
    float bm = 0.f;
    int   bi = -1;
    for (int i = tid; i < K; i += 256) {
      float v = sc[i];
      if (v > bm) { bm = v; bi = i; }  // strict > keeps lowest index on ties
    }
    rs[tid] = bm;
    ri[tid] = bi;
    __syncthreads();
    for (int off = 128; off > 0; off >>= 1) {
      if (tid < off) {
        float s2 = rs[tid + off];
        int   i2 = ri[tid + off];
        if (s2 > rs[tid] || (s2 == rs[tid] && i2 >= 0 && (ri[tid] < 0 || i2 < ri[tid]))) {
          rs[tid] = s2;
          ri[tid] = i2;
        }
      }
      __syncthreads();
    }

    if (tid == 0) {
      int sel = ri[0];
      seli = sel;
      int o = c * MAXO + t;
      if (sel >= 0) {
        float4 b = bx[sel];
        selb = b;
        out[OFF_CID + o]     = (float)(c + 1);
        out[OFF_S + o]       = rs[0];
        out[OFF_B + o * 4 + 0] = b.x;
        out[OFF_B + o * 4 + 1] = b.y;
        out[OFF_B + o * 4 + 2] = b.z;
        out[OFF_B + o * 4 + 3] = b.w;
        out[OFF_OK + o]      = 1.0f;
      } else {
        out[OFF_CID + o]     = 0.f;
        out[OFF_S + o]       = 0.f;
        out[OFF_B + o * 4 + 0] = 0.f;
        out[OFF_B + o * 4 + 1] = 0.f;
        out[OFF_B + o * 4 + 2] = 0.f;
        out[OFF_B + o * 4 + 3] = 0.f;
        out[OFF_OK + o]      = 0.f;
      }
    }
    __syncthreads();

    int si = seli;                 // uniform across block
    if (si < 0) continue;          // nothing left; keep writing zero rows

    float4 B  = selb;
    float  a1 = (B.z - B.x) * (B.w - B.y);
    for (int i = tid; i < K; i += 256) {
      if (sc[i] <= 0.f) continue;  // already suppressed
      float4 bb  = bx[i];
      float yy1 = fmaxf(B.x, bb.x), xx1 = fmaxf(B.y, bb.y);
      float yy2 = fminf(B.z, bb.z), xx2 = fminf(B.w, bb.w);
      float inter = fmaxf(yy2 - yy1, 0.f) * fmaxf(xx2 - xx1, 0.f);
      float a2 = (bb.z - bb.x) * (bb.w - bb.y);
      float iou = inter / (a1 + a2 - inter + 1e-8f);
      if (iou > IOU_THR) sc[i] = 0.f;  // self-IoU == 1 removes `si` too
    }
    __syncthreads();
  }
}

// ------------------------------------------------------------------
extern "C" void kernel_launch(void* const* d_in, const int* in_sizes, int n_in,
                              void* d_out, int out_size, void* d_ws, size_t ws_size,
                              hipStream_t stream) {
  (void)n_in; (void)out_size; (void)ws_size;
  const float* cla = (const float*)d_in[0];
  const float* reg = (const float*)d_in[1];
  const float* anc = (const float*)d_in[2];
  float* out = (float*)d_out;

  // scratch layout: counts | scores | boxes
  char*  ws     = (char*)d_ws;
  int*   counts = (int*)ws;
  float* sws    = (float*)(ws + 4096);
  float4* bws   = (float4*)(ws + 4096 + (size_t)NFG * CAP * sizeof(float));

  const int nAnch  = in_sizes[1] / 4;          // 436,480
  const int blocksA = (nAnch + TILE - 1) / TILE;

  rn_zero_counts<<<1, 128, 0, stream>>>(counts);
  rn_decode_extract<<<blocksA, TILE, 0, stream>>>(cla, reg, anc, sws, bws, counts, nAnch);
  rn_nms<<<NFG, 256, 0, stream>>>(sws, bws, counts, out);
}